// LigerLMHeadKTO_45887430590963
// MI455X (gfx1250) — compile-verified
//
#include <hip/hip_runtime.h>
#include <cstdint>
#include <cstddef>

// Problem constants (match reference)
#define HDIM 4096
#define VDIM 32000
#define TDIM 1024
#define BDIM 2
#define MT 16                             // token rows per tile (WMMA M)
#define NTILES ((BDIM * TDIM) / MT)       // 128 tiles of 16 tokens
#define NCOLS_PER_BLOCK 128               // 8 waves * 16 vocab cols
#define VCHUNKS (VDIM / NCOLS_PER_BLOCK)  // 250
#define PADKF (HDIM + 4)                  // f32 elems per LDS row (+16B pad -> bank spread)
#define IGNORE_INDEX (-100)
#define KTO_BETA 0.1f

typedef float v2f __attribute__((ext_vector_type(2)));
typedef float v8f __attribute__((ext_vector_type(8)));

// Fused linear log-softmax target log-prob with online logsumexp.
// grid = (NTILES, 2): blockIdx.y selects policy (0) / reference (1) path.
// Block: 16 tokens, 8 waves; wave w covers vocab cols (chunk*128 + w*16 + lane&15).
__global__ __launch_bounds__(256)
void kto_logprob_kernel(const float* __restrict__ X0,  // policy activations (B*T, H)
                        const float* __restrict__ X1,  // reference activations
                        const float* __restrict__ W0,  // policy weight (V, H)
                        const float* __restrict__ W1,  // reference weight
                        const int*   __restrict__ Y,   // (B*T,) targets
                        float* __restrict__ part)      // [path][tile] sums, +256: counts
{
    __shared__ __align__(16) float Asf[MT * PADKF];    // 262,400 B f32 A tile
    __shared__ float redM[8 * 16], redS[8 * 16], redT[8 * 16];
    __shared__ float rsum[16], rcnt[16];

    const int tile = blockIdx.x;
    const int path = blockIdx.y;
    const int tid  = threadIdx.x;
    const int lane = tid & 31;
    const int wave = tid >> 5;
    const int half = lane >> 4;     // WMMA lane half: selects K pair / M group
    const int l16  = lane & 15;

    const float* __restrict__ Xp = path ? X1 : X0;
    const float* __restrict__ Wm = path ? W1 : W0;

    // ---- Stage A tile (16 x 4096 f32) into LDS via async global->LDS b128 copies ----
    const float* xrow = Xp + (size_t)tile * MT * HDIM;
    for (int idx = tid; idx < MT * (HDIM / 4); idx += 256) {
        int r  = idx >> 10;                      // HDIM/4 = 1024 x b128 per row
        int k4 = idx & 1023;
        const float* gsrc = xrow + r * HDIM + k4 * 4;
        unsigned int ldsoff = (unsigned int)(size_t)(Asf + r * PADKF + k4 * 4);
        asm volatile("global_load_async_to_lds_b128 %0, %1, off"
                     :: "v"(ldsoff), "v"(gsrc)
                     : "memory");
    }
    asm volatile("s_wait_asynccnt 0x0" ::: "memory");
    __syncthreads();

    // Rows this lane observes in the C/D layout: M = j + 8*half (j = VGPR index)
    int yy[8];
#pragma unroll
    for (int j = 0; j < 8; ++j) yy[j] = Y[tile * MT + j + 8 * half];

    // A operand: M=l16, K pair (k0 + 2*half, +1) -> ds_load_b64 per step
    const float* aL = Asf + l16 * PADKF + 2 * half;
    const float NEG_INF = -__builtin_inff();

    float m[8], s[8], t[8];
#pragma unroll
    for (int j = 0; j < 8; ++j) { m[j] = NEG_INF; s[j] = 0.f; t[j] = 0.f; }

    for (int vc = 0; vc < VCHUNKS; ++vc) {
        const int mycol = vc * NCOLS_PER_BLOCK + wave * 16 + l16;
        // B layout (4x16, KxN): lanes 0-15 hold (K0,K1), lanes 16-31 hold (K2,K3), N=lane&15
        const float* wp = Wm + (size_t)mycol * HDIM + 2 * half;

        // 4 independent accumulators -> 4-deep ILP on the matrix pipe
        v8f acc0 = {0.f,0.f,0.f,0.f,0.f,0.f,0.f,0.f};
        v8f acc1 = {0.f,0.f,0.f,0.f,0.f,0.f,0.f,0.f};
        v8f acc2 = {0.f,0.f,0.f,0.f,0.f,0.f,0.f,0.f};
        v8f acc3 = {0.f,0.f,0.f,0.f,0.f,0.f,0.f,0.f};

        for (int kb = 0; kb < HDIM; kb += 64) {
            __builtin_prefetch(wp + kb + 256, 0, 1);   // global_prefetch ahead of W stream
#pragma unroll
            for (int kk = 0; kk < 64; kk += 16) {
                const int k0 = kb + kk;
                v2f a0 = *(const v2f*)(aL + k0);
                v2f b0 = *(const v2f*)(wp + k0);
                acc0 = __builtin_amdgcn_wmma_f32_16x16x4_f32(
                         false, a0, false, b0, (short)0, acc0, false, false);
                v2f a1 = *(const v2f*)(aL + k0 + 4);
                v2f b1 = *(const v2f*)(wp + k0 + 4);
                acc1 = __builtin_amdgcn_wmma_f32_16x16x4_f32(
                         false, a1, false, b1, (short)0, acc1, false, false);
                v2f a2 = *(const v2f*)(aL + k0 + 8);
                v2f b2 = *(const v2f*)(wp + k0 + 8);
                acc2 = __builtin_amdgcn_wmma_f32_16x16x4_f32(
                         false, a2, false, b2, (short)0, acc2, false, false);
                v2f a3 = *(const v2f*)(aL + k0 + 12);
                v2f b3 = *(const v2f*)(wp + k0 + 12);
                acc3 = __builtin_amdgcn_wmma_f32_16x16x4_f32(
                         false, a3, false, b3, (short)0, acc3, false, false);
            }
        }

        // Online logsumexp per (lane, VGPR-row) + target logit capture
#pragma unroll
        for (int j = 0; j < 8; ++j) {
            float c  = acc0[j] + acc1[j] + acc2[j] + acc3[j];
            float nm = fmaxf(m[j], c);
            s[j] = s[j] * __expf(m[j] - nm) + __expf(c - nm);
            m[j] = nm;
            if (mycol == yy[j]) t[j] += c;
        }
    }

    // ---- Cross-lane merge within each 16-lane half (rows differ between halves) ----
#pragma unroll
    for (int j = 0; j < 8; ++j) {
#pragma unroll
        for (int msk = 1; msk < 16; msk <<= 1) {
            float om = __shfl_xor(m[j], msk, 32);
            float os = __shfl_xor(s[j], msk, 32);
            float ot = __shfl_xor(t[j], msk, 32);
            float nm = fmaxf(m[j], om);
            s[j] = s[j] * __expf(m[j] - nm) + os * __expf(om - nm);
            m[j] = nm;
            t[j] += ot;
        }
    }

    // Lane 0 holds rows 0-7, lane 16 holds rows 8-15 of this wave's vocab slice
    if (l16 == 0) {
#pragma unroll
        for (int j = 0; j < 8; ++j) {
            int row = j + 8 * half;
            redM[wave * 16 + row] = m[j];
            redS[wave * 16 + row] = s[j];
            redT[wave * 16 + row] = t[j];
        }
    }
    __syncthreads();

    // ---- Cross-wave merge: 16 threads, one row each ----
    if (tid < 16) {
        const int row = tid;
        float M = NEG_INF, S = 0.f, T = 0.f;
        for (int w = 0; w < 8; ++w) {
            float om = redM[w * 16 + row];
            float os = redS[w * 16 + row];
            float nm = fmaxf(M, om);
            S = S * __expf(M - nm) + os * __expf(om - nm);
            M = nm;
            T += redT[w * 16 + row];
        }
        float lse = M + __logf(S);
        int   yv  = Y[tile * MT + row];
        float msk = (yv != IGNORE_INDEX) ? 1.f : 0.f;
        rsum[row] = (T - lse) * msk;
        rcnt[row] = msk;
    }
    __syncthreads();

    if (tid == 0) {
        float a = 0.f, c = 0.f;
        for (int r = 0; r < 16; ++r) { a += rsum[r]; c += rcnt[r]; }
        part[path * NTILES + tile]       = a;   // masked sum of token log-probs
        part[256 + path * NTILES + tile] = c;   // mask count
    }
}

// ws layout: [0..127] policy sums, [128..255] ref sums,
//            [256..383] policy counts, [384..511] ref counts
__global__ void kto_finalize(const float* __restrict__ ws,
                             const unsigned char* __restrict__ pref,
                             float* __restrict__ out)
{
    float avg[2][BDIM];
    for (int p = 0; p < 2; ++p)
        for (int b = 0; b < BDIM; ++b) {
            float s = 0.f, c = 0.f;
            const int base = p * NTILES + b * (NTILES / BDIM);
            for (int t = 0; t < NTILES / BDIM; ++t) {
                s += ws[base + t];
                c += ws[256 + base + t];
            }
            avg[p][b] = s / fmaxf(c, 1.f);
        }
    float loss_sum = 0.f;
    for (int b = 0; b < BDIM; ++b) {
        float lr   = avg[0][b] - avg[1][b];
        float mult = pref[b] ? 1.f : -1.f;
        float z    = KTO_BETA * lr * mult;
        loss_sum  += 1.f / (1.f + __expf(z));   // 1 - sigmoid(z)
    }
    *out = loss_sum / (float)BDIM;
}

extern "C" void kernel_launch(void* const* d_in, const int* in_sizes, int n_in,
                              void* d_out, int out_size, void* d_ws, size_t ws_size,
                              hipStream_t stream) {
    (void)in_sizes; (void)n_in; (void)out_size; (void)ws_size;
    const float*         x    = (const float*)d_in[0];
    const float*         rx   = (const float*)d_in[1];
    const int*           y    = (const int*)d_in[2];
    const unsigned char* pref = (const unsigned char*)d_in[3]; // jax bool -> 1 byte
    const float*         W    = (const float*)d_in[4];
    const float*         rW   = (const float*)d_in[5];
    float* ws = (float*)d_ws;

    dim3 grid(NTILES, 2), block(256);
    kto_logprob_kernel<<<grid, block, 0, stream>>>(x, rx, W, rW, y, ws);
    kto_finalize<<<1, 1, 0, stream>>>(ws, pref, (float*)d_out);
}